// Attn_78658031059249
// MI455X (gfx1250) — compile-verified
//
#include <hip/hip_runtime.h>
#include <math.h>

// Problem constants from the reference (B=64, T=2048, H=256).
#define HDIM   256
#define TLEN   2048
#define BATCH  64
#define NEG_INF (-1.0e10f)

typedef __attribute__((ext_vector_type(2))) float v2f;
typedef __attribute__((ext_vector_type(8))) float v8f;

// ---------------------------------------------------------------------------
// Kernel 1: u[h] = sum_k v[k] * W[k,h]   (u = v^T W),  c = dot(b, v).
// Tiny (64K MACs). ws[0..255] = u, ws[256] = c.
// ---------------------------------------------------------------------------
__global__ void __launch_bounds__(HDIM)
precompute_kernel(const float* __restrict__ W,
                  const float* __restrict__ bias,
                  const float* __restrict__ v,
                  float* __restrict__ ws) {
    int h = threadIdx.x;
    float acc = 0.0f;
    // Coalesced: consecutive threads read consecutive addresses of each W row.
    for (int k = 0; k < HDIM; ++k)
        acc = fmaf(v[k], W[k * HDIM + h], acc);
    ws[h] = acc;
    if (h == 0) {
        float c = 0.0f;
        for (int k = 0; k < HDIM; ++k)
            c = fmaf(bias[k], v[k], c);
        ws[HDIM] = c;
    }
}

// ---------------------------------------------------------------------------
// Kernel 2: one block per batch row. Stream Q once (memory-bound: 134 MB
// total -> ~5.8us floor at 23.3 TB/s), compute energies via
// V_WMMA_F32_16X16X4_F32 into LDS, then fused masked softmax over T=2048.
//
// WMMA mapping per wave (wave32, 16 tokens per tile):
//   A (16x4 f32): lane L holds row M=L%16; lanes 0-15 carry K={0,1} in
//                 VGPR{0,1}, lanes 16-31 carry K={2,3}.
//   B (4x16 f32): u[kk..kk+3] broadcast across all 16 columns (same lane
//                 striping as A on K), so every column of D is the energy.
//   D (16x16 f32, 8 VGPRs): VGPR r = M=r (lanes 0-15) / M=8+r (lanes 16-31);
//                 column 0 lives in lanes 0 and 16 -> they write the tile.
// ---------------------------------------------------------------------------
__global__ void __launch_bounds__(512)
attn_energy_softmax_kernel(const float* __restrict__ Q,
                           const int*   __restrict__ lens,
                           const float* __restrict__ ws,
                           float* __restrict__ out) {
    __shared__ float u_lds[HDIM];
    __shared__ float e_lds[TLEN];
    __shared__ float red[512];
    __shared__ float c_sh;

    const int tid = threadIdx.x;
    const int bb  = blockIdx.x;

    if (tid < HDIM) u_lds[tid] = ws[tid];
    if (tid == 0)   c_sh = ws[HDIM];
    __syncthreads();
    const float c = c_sh;

    const int wave = tid >> 5;        // 16 waves
    const int lane = tid & 31;
    const int m    = lane & 15;       // token row within tile
    const int kp   = (lane >> 4) << 1; // K sub-offset: 0 (lanes 0-15) or 2

    const float* qbase = Q + (size_t)bb * TLEN * HDIM;

    // 128 tiles of 16 tokens; wave w handles tiles w, w+16, ... (8 tiles).
    for (int tile = wave; tile < TLEN / 16; tile += 16) {
        const float* qrow = qbase + (size_t)(tile * 16 + m) * HDIM;
        v8f acc = {0.f, 0.f, 0.f, 0.f, 0.f, 0.f, 0.f, 0.f};
#pragma unroll 8
        for (int kk = 0; kk < HDIM; kk += 4) {
            v2f a;
            a.x = qrow[kk + kp];          // contiguous 8B load per lane
            a.y = qrow[kk + kp + 1];
            v2f bv;
            bv.x = u_lds[kk + kp];        // ds_load_b64, same K striping as A
            bv.y = u_lds[kk + kp + 1];
            // (neg_a, A, neg_b, B, c_mod, C, reuse_a, reuse_b)
            acc = __builtin_amdgcn_wmma_f32_16x16x4_f32(
                false, a, false, bv, (short)0, acc, false, false);
        }
        // Column 0 of D: lane 0 holds M=0..7 (VGPR 0..7), lane 16 holds M=8..15.
        if ((lane & 15) == 0) {
            const int mbase = tile * 16 + (lane >> 4) * 8;
#pragma unroll
            for (int r = 0; r < 8; ++r)
                e_lds[mbase + r] = acc[r] + c;
        }
    }
    __syncthreads();

    // ---------------- masked softmax over T=2048 ----------------
    const int len = lens[bb];
    float vals[4];
    float lm = NEG_INF;
#pragma unroll
    for (int i = 0; i < 4; ++i) {
        const int tok = tid + i * 512;
        const float e = (tok < len) ? e_lds[tok] : NEG_INF;
        vals[i] = e;
        lm = fmaxf(lm, e);
    }
    red[tid] = lm;
    __syncthreads();
    for (int s = 256; s > 0; s >>= 1) {
        if (tid < s) red[tid] = fmaxf(red[tid], red[tid + s]);
        __syncthreads();
    }
    const float mx = red[0];
    __syncthreads();

    float pv[4];
    float ls = 0.0f;
#pragma unroll
    for (int i = 0; i < 4; ++i) {
        pv[i] = expf(vals[i] - mx);   // masked tokens: expf(-1e10-mx) == 0.0f
        ls += pv[i];
    }
    red[tid] = ls;
    __syncthreads();
    for (int s = 256; s > 0; s >>= 1) {
        if (tid < s) red[tid] += red[tid + s];
        __syncthreads();
    }
    const float inv = 1.0f / red[0];

    float* orow = out + (size_t)bb * TLEN;
#pragma unroll
    for (int i = 0; i < 4; ++i) {
        const int tok = tid + i * 512;
        orow[tok] = pv[i] * inv;      // coalesced stores
    }
}

// ---------------------------------------------------------------------------
// Host-side launcher.
// Inputs (setup_inputs order): questions f32[B,T,H], questions_lens i32[B],
// W f32[H,H], b f32[H], v f32[H]. Output: f32[B,T].
// ---------------------------------------------------------------------------
extern "C" void kernel_launch(void* const* d_in, const int* in_sizes, int n_in,
                              void* d_out, int out_size, void* d_ws, size_t ws_size,
                              hipStream_t stream) {
    const float* Q    = (const float*)d_in[0];
    const int*   lens = (const int*)  d_in[1];
    const float* W    = (const float*)d_in[2];
    const float* bias = (const float*)d_in[3];
    const float* v    = (const float*)d_in[4];
    float* out = (float*)d_out;
    float* ws  = (float*)d_ws;   // ws[0..255] = u = v^T W, ws[256] = b.v

    precompute_kernel<<<1, HDIM, 0, stream>>>(W, bias, v, ws);
    attn_energy_softmax_kernel<<<BATCH, 512, 0, stream>>>(Q, lens, ws, out);
}